// PatchedGroup_85942295593337
// MI455X (gfx1250) — compile-verified
//
#include <hip/hip_runtime.h>
#include <hip/hip_bf16.h>

typedef __attribute__((ext_vector_type(2))) float v2f;
typedef __attribute__((ext_vector_type(8))) float v8f;

#define NPTS   8192
#define NGRP   512
#define GSIZE  32
#define NBATCH 8
#define THREADS 512      // 16 waves of 32
#define PPT    16        // points per thread (8192/512)
#define GT     8         // groups per knn block (LDS: 8*8192*4 = 256KB <= 320KB)

// ---------------------------------------------------------------------------
// Kernel 1: farthest point sampling. One block per batch, points in registers.
// ---------------------------------------------------------------------------
__global__ void fps_kernel(const float* __restrict__ xyz,
                           float* __restrict__ centersWs,
                           float* __restrict__ centerOut) {
  __shared__ float redV[16];
  __shared__ int   redI[16];
  __shared__ int   winner;
  __shared__ float bc[3];

  const int tid  = threadIdx.x;
  const int lane = tid & 31;
  const int wave = tid >> 5;
  const int b    = blockIdx.x;
  const float* pb = xyz + (size_t)b * NPTS * 6;

  float px[PPT], py[PPT], pz[PPT], dist[PPT];
#pragma unroll
  for (int t = 0; t < PPT; ++t) {
    const float* p = pb + (size_t)(tid + t * THREADS) * 6;
    px[t] = p[0]; py[t] = p[1]; pz[t] = p[2];
  }

  if (tid == 0) {
    bc[0] = pb[0]; bc[1] = pb[1]; bc[2] = pb[2];
    float* cw = centersWs + (size_t)(b * NGRP) * 3;
    float* co = centerOut + (size_t)(b * NGRP) * 3;
    cw[0] = pb[0]; cw[1] = pb[1]; cw[2] = pb[2];
    co[0] = pb[0]; co[1] = pb[1]; co[2] = pb[2];
  }
  __syncthreads();

  float fx = bc[0], fy = bc[1], fz = bc[2];
#pragma unroll
  for (int t = 0; t < PPT; ++t) {
    float dx = px[t] - fx, dy = py[t] - fy, dz = pz[t] - fz;
    dist[t] = dx * dx + dy * dy + dz * dz;
  }

  for (int s = 1; s < NGRP; ++s) {
    // thread-local argmax (lowest index wins ties: ascending t, strict >)
    float bv = dist[0]; int bi = tid;
#pragma unroll
    for (int t = 1; t < PPT; ++t) {
      if (dist[t] > bv) { bv = dist[t]; bi = tid + t * THREADS; }
    }
    // wave32 reduction
    float v = bv; int i = bi;
#pragma unroll
    for (int off = 16; off > 0; off >>= 1) {
      float ov = __shfl_down(v, off);
      int   oi = __shfl_down(i, off);
      if (ov > v || (ov == v && oi < i)) { v = ov; i = oi; }
    }
    if (lane == 0) { redV[wave] = v; redI[wave] = i; }
    __syncthreads();
    if (tid == 0) {
      float fv = redV[0]; int fi = redI[0];
      for (int w2 = 1; w2 < 16; ++w2)
        if (redV[w2] > fv || (redV[w2] == fv && redI[w2] < fi)) { fv = redV[w2]; fi = redI[w2]; }
      winner = fi;
    }
    __syncthreads();
    int w = winner;
    if ((w & (THREADS - 1)) == tid) {
      int tsel = w >> 9;
      float sx = 0.f, sy = 0.f, sz = 0.f;
#pragma unroll
      for (int j = 0; j < PPT; ++j)
        if (j == tsel) { sx = px[j]; sy = py[j]; sz = pz[j]; }
      bc[0] = sx; bc[1] = sy; bc[2] = sz;
      float* cw = centersWs + (size_t)(b * NGRP + s) * 3;
      float* co = centerOut + (size_t)(b * NGRP + s) * 3;
      cw[0] = sx; cw[1] = sy; cw[2] = sz;
      co[0] = sx; co[1] = sy; co[2] = sz;
    }
    __syncthreads();
    fx = bc[0]; fy = bc[1]; fz = bc[2];
#pragma unroll
    for (int t = 0; t < PPT; ++t) {
      float dx = px[t] - fx, dy = py[t] - fy, dz = pz[t] - fz;
      float dd = dx * dx + dy * dy + dz * dz;
      dist[t] = fminf(dist[t], dd);
    }
  }
}

// ---------------------------------------------------------------------------
// Kernel 2: d2 via V_WMMA_F32_16X16X4_F32 (A=[-2c, |c|^2], B=[p, 1]) staged to
// LDS, then 32 block-argmin extraction rounds per group. 8 groups per block.
// ---------------------------------------------------------------------------
__global__ void knn_kernel(const float* __restrict__ xyz,
                           const float* __restrict__ centersWs,
                           int* __restrict__ idxWs) {
  extern __shared__ float smem[];
  float* sd2  = smem;                       // GT * NPTS floats
  float* redV = smem + GT * NPTS;           // 16
  int*   redI = (int*)(redV + 16);          // 16
  int*   outI = (int*)(redI + 16);          // 1

  const int tid   = threadIdx.x;
  const int lane  = tid & 31;
  const int wave  = tid >> 5;
  const int b     = blockIdx.x >> 6;                 // 64 tiles per batch
  const int gbase = (blockIdx.x & 63) * GT;

  // Build A fragment (16x4 fp32: lanes 0-15 hold K=0,1; lanes 16-31 hold K=2,3)
  int m = lane & 15;
  float cx = 0.f, cy = 0.f, cz = 0.f;
  if (m < GT) {
    const float* c = centersWs + (size_t)(b * NGRP + gbase + m) * 3;
    cx = c[0]; cy = c[1]; cz = c[2];
  }
  v2f a;
  if (lane < 16) { a.x = -2.0f * cx; a.y = -2.0f * cy; }
  else           { a.x = -2.0f * cz; a.y = cx * cx + cy * cy + cz * cz; }

  const float* pb = xyz + (size_t)b * NPTS * 6;

  // Phase 1: each wave computes its share of 16-point chunks via WMMA
  for (int it = 0; it < 32; ++it) {
    int chunk = wave + it * 16;
    int n = chunk * 16 + (lane & 15);
    const float* p = pb + (size_t)n * 6;
    float x = p[0], y = p[1], z = p[2];
    float pn2 = x * x + y * y + z * z;
    v2f bb;
    if (lane < 16) { bb.x = x; bb.y = y; }
    else           { bb.x = z; bb.y = 1.0f; }
    v8f acc = {};
    acc = __builtin_amdgcn_wmma_f32_16x16x4_f32(false, a, false, bb,
                                                (short)0, acc, false, false);
    if (lane < 16) {
#pragma unroll
      for (int i = 0; i < GT; ++i)            // rows 0..7 live in lanes 0-15
        sd2[i * NPTS + n] = acc[i] + pn2;
    }
  }
  __syncthreads();

  // Phase 2: per group, 32 rounds of block-wide argmin (lowest index on ties)
  for (int g = 0; g < GT; ++g) {
    float d[PPT];
#pragma unroll
    for (int t = 0; t < PPT; ++t) d[t] = sd2[g * NPTS + tid + t * THREADS];

    for (int k = 0; k < GSIZE; ++k) {
      float bv = d[0]; int bi = tid;
#pragma unroll
      for (int t = 1; t < PPT; ++t) {
        if (d[t] < bv) { bv = d[t]; bi = tid + t * THREADS; }
      }
      float v = bv; int i = bi;
#pragma unroll
      for (int off = 16; off > 0; off >>= 1) {
        float ov = __shfl_down(v, off);
        int   oi = __shfl_down(i, off);
        if (ov < v || (ov == v && oi < i)) { v = ov; i = oi; }
      }
      if (lane == 0) { redV[wave] = v; redI[wave] = i; }
      __syncthreads();
      if (tid == 0) {
        float fv = redV[0]; int fi = redI[0];
        for (int w2 = 1; w2 < 16; ++w2)
          if (redV[w2] < fv || (redV[w2] == fv && redI[w2] < fi)) { fv = redV[w2]; fi = redI[w2]; }
        *outI = fi;
        idxWs[(size_t)(b * NGRP + gbase + g) * GSIZE + k] = fi;
      }
      __syncthreads();
      int w = *outI;
      if ((w & (THREADS - 1)) == tid) {
        int tsel = w >> 9;
#pragma unroll
        for (int j = 0; j < PPT; ++j)
          if (j == tsel) d[j] = 3.0e38f;      // remove winner
      }
    }
    __syncthreads();
  }
}

// ---------------------------------------------------------------------------
// Kernel 3: gather neighborhoods (rel xyz + extra channels) and idx as float.
// ---------------------------------------------------------------------------
__global__ void gather_kernel(const float* __restrict__ xyz,
                              const float* __restrict__ centersWs,
                              const int* __restrict__ idxWs,
                              float* __restrict__ outNeigh,
                              float* __restrict__ outIdx) {
  int lin = blockIdx.x * 256 + threadIdx.x;
  if (lin >= NBATCH * NGRP * GSIZE) return;
  int g = (lin >> 5) & (NGRP - 1);
  int b = lin >> 14;
  int idx = idxWs[lin];
  const float* p = xyz + ((size_t)b * NPTS + idx) * 6;
  const float* c = centersWs + (size_t)(b * NGRP + g) * 3;
  float* o = outNeigh + (size_t)lin * 6;
  o[0] = p[0] - c[0];
  o[1] = p[1] - c[1];
  o[2] = p[2] - c[2];
  o[3] = p[3];
  o[4] = p[4];
  o[5] = p[5];
  outIdx[lin] = (float)idx;
}

// ---------------------------------------------------------------------------
extern "C" void kernel_launch(void* const* d_in, const int* in_sizes, int n_in,
                              void* d_out, int out_size, void* d_ws, size_t ws_size,
                              hipStream_t stream) {
  (void)in_sizes; (void)n_in; (void)out_size; (void)ws_size;
  const float* xyz = (const float*)d_in[0];
  float* out = (float*)d_out;

  // output layout: neighborhood [8,512,32,6] | center [8,512,3] | idx [8,512,32]
  float* outNeigh  = out;                                   // 786432 floats
  float* outCenter = out + (size_t)NBATCH * NGRP * GSIZE * 6;   // +12288
  float* outIdx    = outCenter + (size_t)NBATCH * NGRP * 3;     // +131072

  float* centersWs = (float*)d_ws;                              // 12288 floats
  int*   idxWs     = (int*)((char*)d_ws + (size_t)NBATCH * NGRP * 3 * sizeof(float));

  fps_kernel<<<NBATCH, THREADS, 0, stream>>>(xyz, centersWs, outCenter);

  size_t shmem = (size_t)GT * NPTS * sizeof(float) + 256;   // 256KB + scratch
  knn_kernel<<<NBATCH * (NGRP / GT), THREADS, shmem, stream>>>(xyz, centersWs, idxWs);

  int total = NBATCH * NGRP * GSIZE;
  gather_kernel<<<(total + 255) / 256, 256, 0, stream>>>(xyz, centersWs, idxWs,
                                                         outNeigh, outIdx);
}